// GINEncoder_27333171871744
// MI455X (gfx1250) — compile-verified
//
#include <hip/hip_runtime.h>
#include <hip/hip_bf16.h>

typedef __attribute__((ext_vector_type(2))) float v2f;
typedef __attribute__((ext_vector_type(8))) float v8f;

#define NGRAPHS 256
#define HIDC    128
#define BN_EPS  1e-5f

// ---------------------------------------------------------------------------
// Utility kernels
// ---------------------------------------------------------------------------
__global__ __launch_bounds__(256) void zero_kernel(float* __restrict__ p, long n) {
  long i = (long)blockIdx.x * blockDim.x + threadIdx.x;
  if (i < n) p[i] = 0.0f;
}

__global__ __launch_bounds__(256) void copy4_kernel(const float* __restrict__ src,
                                                    float* __restrict__ dst, long n4) {
  long i = (long)blockIdx.x * blockDim.x + threadIdx.x;
  if (i < n4) ((float4*)dst)[i] = ((const float4*)src)[i];
}

// ---------------------------------------------------------------------------
// Edge aggregation: acc[dst[e]][c] += h[src[e]][c]   (acc pre-initialized = h)
// One thread handles (edge, 4 channels); 16B gather + 4 fp32 atomics.
// Node table (51MB) fits in 192MB L2 -> gathers/atomics mostly L2-resident.
// ---------------------------------------------------------------------------
__global__ __launch_bounds__(256) void scatter_add_kernel(
    const float* __restrict__ h, const int* __restrict__ src,
    const int* __restrict__ dst, float* __restrict__ acc,
    int nEdges, int C) {
  const int cpt = C >> 2;  // threads per edge
  long t = (long)blockIdx.x * blockDim.x + threadIdx.x;
  long total = (long)nEdges * cpt;
  if (t >= total) return;
  int e  = (int)(t / cpt);
  int ci = (int)(t - (long)e * cpt) << 2;
  int s = src[e], d = dst[e];
  float4 v = *(const float4*)(h + (long)s * C + ci);
  float* p = acc + (long)d * C + ci;
  atomicAdd(p + 0, v.x);
  atomicAdd(p + 1, v.y);
  atomicAdd(p + 2, v.z);
  atomicAdd(p + 3, v.w);
}

// ---------------------------------------------------------------------------
// GEMM: C[M,128] = A[M,K] * W[K,128] (+bias, optional ReLU), K in {64,128}.
// Block = 256 thr = 8 waves. Block owns a 16-row M tile; wave w owns N tile
// [16w, 16w+16). A tile staged in LDS (stride K+1 -> conflict free).
// K-loop of v_wmma_f32_16x16x4_f32 (full fp32, matches fp32 reference).
// ---------------------------------------------------------------------------
__global__ __launch_bounds__(256) void gemm_bias_kernel(
    const float* __restrict__ A, const float* __restrict__ W,
    const float* __restrict__ bias, float* __restrict__ C,
    int M, int K, int relu) {
  __shared__ float As[16 * 132];  // max K=128 -> stride 129 used
  const int tid  = threadIdx.x;
  const int m0   = blockIdx.x << 4;
  const int lane = tid & 31;
  const int wave = tid >> 5;
  const int nbase = wave << 4;
  const int sA = K + 1;

  // cooperative A-tile load (16 x K)
  for (int idx = tid; idx < 16 * K; idx += 256) {
    int r = idx / K, c = idx - r * K;
    int gm = m0 + r;
    As[r * sA + c] = (gm < M) ? A[(long)gm * K + c] : 0.0f;
  }
  __syncthreads();

  // fragment coordinates (ISA 7.12.2):
  // A 16x4: lanes 0-15 hold M=lane, K pair {0,1}; lanes 16-31 K pair {2,3}
  // B 4x16 / C 16x16: N = lane&15; lane-half selects K/M sub-block.
  const int row = lane & 15;
  const int kb  = (lane >> 4) << 1;
  const int n   = nbase + (lane & 15);

  v8f acc = {};
  for (int k0 = 0; k0 < K; k0 += 4) {
    v2f a, b;
    a.x = As[row * sA + k0 + kb];
    a.y = As[row * sA + k0 + kb + 1];
    b.x = W[(long)(k0 + kb)     * HIDC + n];
    b.y = W[(long)(k0 + kb + 1) * HIDC + n];
    acc = __builtin_amdgcn_wmma_f32_16x16x4_f32(
        /*neg_a=*/false, a, /*neg_b=*/false, b,
        /*c_mod=*/(short)0, acc, /*reuse_a=*/false, /*reuse_b=*/false);
  }

  const float bv = bias[n];
  const int mhalf = m0 + ((lane >> 4) << 3);
#pragma unroll
  for (int r = 0; r < 8; ++r) {
    int m = mhalf + r;  // C layout: VGPR r -> M = r (+8 for upper lane half)
    if (m < M) {
      float v = acc[r] + bv;
      if (relu) v = fmaxf(v, 0.0f);
      C[(long)m * HIDC + n] = v;
    }
  }
}

// ---------------------------------------------------------------------------
// BatchNorm: per-channel sum / sumsq over all rows (grid-stride + atomics)
// ---------------------------------------------------------------------------
__global__ __launch_bounds__(256) void bn_stats_kernel(
    const float* __restrict__ z, float* __restrict__ sums,
    float* __restrict__ sumsq, int nRows) {
  int gtid = blockIdx.x * blockDim.x + threadIdx.x;
  int c  = gtid & (HIDC - 1);
  int r0 = gtid >> 7;
  int rs = (gridDim.x * blockDim.x) >> 7;
  float s = 0.0f, q = 0.0f;
  for (int r = r0; r < nRows; r += rs) {
    float v = z[(long)r * HIDC + c];
    s += v;
    q += v * v;
  }
  atomicAdd(&sums[c], s);
  atomicAdd(&sumsq[c], q);
}

__global__ __launch_bounds__(128) void bn_finalize_kernel(
    const float* __restrict__ sums, const float* __restrict__ sumsq,
    const float* __restrict__ gamma, const float* __restrict__ beta,
    float* __restrict__ scale, float* __restrict__ shift, float invN) {
  int c = threadIdx.x;
  float mean = sums[c] * invN;
  float var  = sumsq[c] * invN - mean * mean;
  float sc   = gamma[c] * rsqrtf(var + BN_EPS);
  scale[c] = sc;
  shift[c] = beta[c] - mean * sc;
}

// h = relu(z*scale + shift); written to hout AND back into z in place so z
// doubles as the next layer's aggregation accumulator (identity term of GIN).
__global__ __launch_bounds__(256) void bn_apply_kernel(
    float* __restrict__ z, float* __restrict__ hout,
    const float* __restrict__ scale, const float* __restrict__ shift, long n) {
  long i = (long)blockIdx.x * blockDim.x + threadIdx.x;
  if (i >= n) return;
  int c = (int)(i & (HIDC - 1));
  float v = fmaxf(z[i] * scale[c] + shift[c], 0.0f);
  hout[i] = v;
  z[i]    = v;
}

// ---------------------------------------------------------------------------
// Graph mean pool
// ---------------------------------------------------------------------------
__global__ __launch_bounds__(256) void pool_sum_kernel(
    const float* __restrict__ h, const int* __restrict__ batch,
    float* __restrict__ pool, int nRows) {
  long t = (long)blockIdx.x * blockDim.x + threadIdx.x;
  int r  = (int)(t >> 5);
  int ci = (int)(t & 31) << 2;
  if (r >= nRows) return;
  int g = batch[r];
  float4 v = *(const float4*)(h + (long)r * HIDC + ci);
  float* p = pool + (long)g * HIDC + ci;
  atomicAdd(p + 0, v.x);
  atomicAdd(p + 1, v.y);
  atomicAdd(p + 2, v.z);
  atomicAdd(p + 3, v.w);
}

__global__ __launch_bounds__(256) void pool_cnt_kernel(
    const int* __restrict__ batch, float* __restrict__ cnt, int nRows) {
  int r = blockIdx.x * blockDim.x + threadIdx.x;
  if (r < nRows) atomicAdd(&cnt[batch[r]], 1.0f);
}

__global__ __launch_bounds__(256) void pool_div_kernel(
    const float* __restrict__ pool, const float* __restrict__ cnt,
    float* __restrict__ hg) {
  int i = blockIdx.x * blockDim.x + threadIdx.x;
  if (i >= NGRAPHS * HIDC) return;
  float c = fmaxf(cnt[i >> 7], 1.0f);
  hg[i] = pool[i] / c;
}

// ---------------------------------------------------------------------------
// Host orchestration
// ---------------------------------------------------------------------------
extern "C" void kernel_launch(void* const* d_in, const int* in_sizes, int n_in,
                              void* d_out, int out_size, void* d_ws, size_t ws_size,
                              hipStream_t stream) {
  const float* x     = (const float*)d_in[0];
  const int*   eidx  = (const int*)d_in[1];
  const int*   batch = (const int*)d_in[2];
  const int nEdges = in_sizes[1] / 2;
  const int nNodes = in_sizes[2];
  const int inC    = in_sizes[0] / nNodes;  // 64
  const int* srcI = eidx;
  const int* dstI = eidx + nEdges;

  const float *w1[3], *b1[3], *w2[3], *b2[3], *gm[3], *bt[3];
  for (int l = 0; l < 3; ++l) {
    w1[l] = (const float*)d_in[3 + 6 * l];
    b1[l] = (const float*)d_in[4 + 6 * l];
    w2[l] = (const float*)d_in[5 + 6 * l];
    b2[l] = (const float*)d_in[6 + 6 * l];
    gm[l] = (const float*)d_in[7 + 6 * l];
    bt[l] = (const float*)d_in[8 + 6 * l];
  }
  const float* projW = (const float*)d_in[21];
  const float* projB = (const float*)d_in[22];

  // ---- workspace carve-up ----
  char* ws = (char*)d_ws;
  size_t off = 0;
  auto carve = [&](size_t bytes) -> float* {
    float* p = (float*)(ws + off);
    off = (off + bytes + 255) & ~(size_t)255;
    return p;
  };
  const size_t featB = (size_t)nNodes * HIDC * sizeof(float);
  float* bufA  = carve(featB);                 // h (layer output)
  float* bufB  = carve(featB);                 // z / aggregation accumulator
  float* bufC  = carve(featB);                 // MLP intermediate
  float* sums  = carve(HIDC * sizeof(float));
  float* sumsq = carve(HIDC * sizeof(float));
  float* scale = carve(HIDC * sizeof(float));
  float* shift = carve(HIDC * sizeof(float));
  float* pool  = carve((size_t)NGRAPHS * HIDC * sizeof(float));
  float* cnt   = carve(NGRAPHS * sizeof(float));
  float* hg    = carve((size_t)NGRAPHS * HIDC * sizeof(float));

  const int T = 256;
  const long nFeat = (long)nNodes * HIDC;
  const int gemmBlocks = (nNodes + 15) / 16;
  const float invN = 1.0f / (float)nNodes;

  for (int l = 0; l < 3; ++l) {
    const int K = (l == 0) ? inC : HIDC;
    const float* hin = (l == 0) ? x : bufA;

    if (l == 0) {  // layers 1,2: bn_apply already left bufB == h
      long n4 = ((long)nNodes * K) / 4;
      copy4_kernel<<<(int)((n4 + T - 1) / T), T, 0, stream>>>(x, bufB, n4);
    }
    // aggregation: bufB[d] += hin[s]
    {
      long tot = (long)nEdges * (K >> 2);
      scatter_add_kernel<<<(int)((tot + T - 1) / T), T, 0, stream>>>(
          hin, srcI, dstI, bufB, nEdges, K);
    }
    // MLP: bufC = relu(bufB @ w1 + b1); bufB = bufC @ w2 + b2
    gemm_bias_kernel<<<gemmBlocks, T, 0, stream>>>(bufB, w1[l], b1[l], bufC,
                                                   nNodes, K, 1);
    gemm_bias_kernel<<<gemmBlocks, T, 0, stream>>>(bufC, w2[l], b2[l], bufB,
                                                   nNodes, HIDC, 0);
    // BatchNorm (training stats) + ReLU
    zero_kernel<<<1, T, 0, stream>>>(sums, 2 * HIDC);  // sums & sumsq adjacent
    bn_stats_kernel<<<512, T, 0, stream>>>(bufB, sums, sumsq, nNodes);
    bn_finalize_kernel<<<1, HIDC, 0, stream>>>(sums, sumsq, gm[l], bt[l],
                                               scale, shift, invN);
    bn_apply_kernel<<<(int)((nFeat + T - 1) / T), T, 0, stream>>>(
        bufB, bufA, scale, shift, nFeat);
  }

  // global mean pool + projection
  zero_kernel<<<((NGRAPHS * HIDC + NGRAPHS) + T - 1) / T, T, 0, stream>>>(
      pool, NGRAPHS * HIDC + NGRAPHS);  // pool & cnt adjacent
  {
    long tot = (long)nNodes * 32;
    pool_sum_kernel<<<(int)((tot + T - 1) / T), T, 0, stream>>>(bufA, batch,
                                                                pool, nNodes);
  }
  pool_cnt_kernel<<<(nNodes + T - 1) / T, T, 0, stream>>>(batch, cnt, nNodes);
  pool_div_kernel<<<(NGRAPHS * HIDC + T - 1) / T, T, 0, stream>>>(pool, cnt, hg);
  gemm_bias_kernel<<<NGRAPHS / 16, T, 0, stream>>>(hg, projW, projB,
                                                   (float*)d_out, NGRAPHS, HIDC, 0);
}